// AlignmentHead_82317343195893
// MI455X (gfx1250) — compile-verified
//
#include <hip/hip_runtime.h>
#include <math.h>

#define NMS_IOU   0.5f
#define MERGE_IOU 0.7f
#define FEPS      1e-8f
#define TWO_PI_F  6.28318530717958647692f
#define NBOX      1024

typedef __attribute__((ext_vector_type(2))) float v2f;
typedef __attribute__((ext_vector_type(8))) float v8f;

// ---------------------------------------------------------------------------
// Rotated-rectangle intersection area, faithful to the reference:
// 24 candidate points (4+4 corners, 16 edge intersections), mask, centroid,
// angular sort, clamp-replicate invalid tail, shoelace, gate on m>=3.
// Fully unrolled so all array indices are compile-time constants (registers).
// ---------------------------------------------------------------------------
__device__ __forceinline__ float pair_inter_area(
    float cx1, float cy1, float w1, float l1, float a1,
    float cx2, float cy2, float w2, float l2, float a2)
{
  const float sxk[4] = {0.5f, 0.5f, -0.5f, -0.5f};
  const float syk[4] = {0.5f, -0.5f, -0.5f, 0.5f};
  float cs1 = cosf(a1), sn1 = sinf(a1);
  float cs2 = cosf(a2), sn2 = sinf(a2);

  float c1x[4], c1y[4], c2x[4], c2y[4];
#pragma unroll
  for (int k = 0; k < 4; ++k) {
    float ox = sxk[k] * w1, oy = syk[k] * l1;
    c1x[k] = cx1 + ox * cs1 - oy * sn1;
    c1y[k] = cy1 + ox * sn1 + oy * cs1;
    ox = sxk[k] * w2; oy = syk[k] * l2;
    c2x[k] = cx2 + ox * cs2 - oy * sn2;
    c2y[k] = cy2 + ox * sn2 + oy * cs2;
  }

  float px[24], py[24];
  bool  mk[24];
#pragma unroll
  for (int k = 0; k < 4; ++k) {
    // corners of box1 inside box2
    float dx = c1x[k] - cx2, dy = c1y[k] - cy2;
    float u = cs2 * dx + sn2 * dy;
    float v = -sn2 * dx + cs2 * dy;
    px[k] = c1x[k]; py[k] = c1y[k];
    mk[k] = (fabsf(u) <= 0.5f * w2 + FEPS) && (fabsf(v) <= 0.5f * l2 + FEPS);
    // corners of box2 inside box1
    dx = c2x[k] - cx1; dy = c2y[k] - cy1;
    u = cs1 * dx + sn1 * dy;
    v = -sn1 * dx + cs1 * dy;
    px[4 + k] = c2x[k]; py[4 + k] = c2y[k];
    mk[4 + k] = (fabsf(u) <= 0.5f * w1 + FEPS) && (fabsf(v) <= 0.5f * l1 + FEPS);
  }

  // 16 segment-segment intersections (edges of box1 x edges of box2)
#pragma unroll
  for (int ii = 0; ii < 4; ++ii) {
    float pxe = c1x[ii], pye = c1y[ii];
    float rx = c1x[(ii + 1) & 3] - pxe, ry = c1y[(ii + 1) & 3] - pye;
#pragma unroll
    for (int jj = 0; jj < 4; ++jj) {
      float qx = c2x[jj], qy = c2y[jj];
      float ux = c2x[(jj + 1) & 3] - qx, uy = c2y[(jj + 1) & 3] - qy;
      float d = rx * uy - ry * ux;
      bool dok = fabsf(d) > FEPS;
      float dn = dok ? d : 1.0f;
      float qpx = qx - pxe, qpy = qy - pye;
      float t = (qpx * uy - qpy * ux) / dn;
      float s = (qpx * ry - qpy * rx) / dn;
      int o = 8 + ii * 4 + jj;
      px[o] = pxe + t * rx;
      py[o] = pye + t * ry;
      mk[o] = dok && (t >= 0.f) && (t <= 1.f) && (s >= 0.f) && (s <= 1.f);
    }
  }

  int m = 0; float sxs = 0.f, sys = 0.f;
#pragma unroll
  for (int k = 0; k < 24; ++k) {
    m   += mk[k] ? 1 : 0;
    sxs += mk[k] ? px[k] : 0.f;
    sys += mk[k] ? py[k] : 0.f;
  }
  float invm = 1.0f / (float)(m > 1 ? m : 1);
  float cenx = sxs * invm, ceny = sys * invm;

  float ag[24];
#pragma unroll
  for (int k = 0; k < 24; ++k)
    ag[k] = mk[k] ? atan2f(py[k] - ceny, px[k] - cenx) : 1.0e9f;

  // odd-even transposition sort (24 passes, constant indices -> registers)
#pragma unroll
  for (int pass = 0; pass < 24; ++pass) {
#pragma unroll
    for (int i = (pass & 1); i < 23; i += 2) {
      if (ag[i] > ag[i + 1]) {
        float t0 = ag[i]; ag[i] = ag[i + 1]; ag[i + 1] = t0;
        t0 = px[i]; px[i] = px[i + 1]; px[i + 1] = t0;
        t0 = py[i]; py[i] = py[i + 1]; py[i + 1] = t0;
      }
    }
  }
  // replicate last valid point over the invalid tail (== ref's idx clamp)
#pragma unroll
  for (int k = 1; k < 24; ++k) {
    bool bad = ag[k] > 9.0e8f;
    px[k] = bad ? px[k - 1] : px[k];
    py[k] = bad ? py[k - 1] : py[k];
  }
  float a2x = 0.f;
#pragma unroll
  for (int k = 0; k < 24; ++k) {
    int n2 = (k + 1) % 24;
    a2x += px[k] * py[n2] - px[n2] * py[k];
  }
  float area = 0.5f * fabsf(a2x);
  return (m >= 3) ? area : 0.0f;
}

// ---------------------------------------------------------------------------
// K1: sigmoid scores
// ---------------------------------------------------------------------------
__global__ void score_kernel(const float* __restrict__ cls,
                             float* __restrict__ scores, int n)
{
  int i = blockIdx.x * blockDim.x + threadIdx.x;
  if (i < n) scores[i] = 1.0f / (1.0f + expf(-cls[i]));
}

// ---------------------------------------------------------------------------
// K2: full N x N rotated IoU per batch, with bounding-circle quick reject.
// ---------------------------------------------------------------------------
__global__ void iou_kernel(const float* __restrict__ anchors,
                           float* __restrict__ iou, int B, int N)
{
  size_t idx = (size_t)blockIdx.x * blockDim.x + threadIdx.x;
  size_t total = (size_t)B * N * N;
  if (idx >= total) return;
  int b = (int)(idx / ((size_t)N * N));
  int r = (int)(idx % ((size_t)N * N));
  int i = r / N, j = r % N;

  const float* A = anchors + ((size_t)b * N + i) * 7;
  const float* C = anchors + ((size_t)b * N + j) * 7;
  float cx1 = A[0], cy1 = A[1], w1 = A[3], l1 = A[4], a1 = A[6];
  float cx2 = C[0], cy2 = C[1], w2 = C[3], l2 = C[4], a2 = C[6];
  float ar1 = w1 * l1, ar2 = w2 * l2;

  float dx = cx1 - cx2, dy = cy1 - cy2;
  float rad = 0.5f * (sqrtf(w1 * w1 + l1 * l1) + sqrtf(w2 * w2 + l2 * l2));
  float inter = 0.0f;
  if (dx * dx + dy * dy <= rad * rad)
    inter = pair_inter_area(cx1, cy1, w1, l1, a1, cx2, cy2, w2, l2, a2);

  iou[idx] = inter / fmaxf(ar1 + ar2 - inter, FEPS);
}

// ---------------------------------------------------------------------------
// K3: bitonic argsort of -score (ascending) => descending score order.
// One 1024-thread workgroup per batch; keys+values in LDS.
// ---------------------------------------------------------------------------
__global__ void sort_kernel(const float* __restrict__ scores,
                            int* __restrict__ order, int N)
{
  __shared__ float key[NBOX];
  __shared__ int   val[NBOX];
  int t = threadIdx.x;
  int b = blockIdx.x;
  key[t] = -scores[(size_t)b * N + t];
  val[t] = t;
  __syncthreads();
  for (int size = 2; size <= N; size <<= 1) {
    for (int stride = size >> 1; stride > 0; stride >>= 1) {
      int ixj = t ^ stride;
      if (ixj > t) {
        bool asc = ((t & size) == 0);
        float ka = key[t], kb = key[ixj];
        bool sw = asc ? (ka > kb) : (ka < kb);
        if (sw) {
          key[t] = kb; key[ixj] = ka;
          int va = val[t]; val[t] = val[ixj]; val[ixj] = va;
        }
      }
      __syncthreads();
    }
  }
  order[(size_t)b * N + t] = val[t];
}

// ---------------------------------------------------------------------------
// K4: sequential NMS scan in sorted space. Writes at iteration i only touch
// sup[t] with t > i, so reading sup[i] is race-free with one barrier/iter.
// ---------------------------------------------------------------------------
__global__ void nms_kernel(const float* __restrict__ iou,
                           const int* __restrict__ order,
                           int* __restrict__ keep, int N)
{
  __shared__ int sup[NBOX];
  __shared__ int ord[NBOX];
  int t = threadIdx.x;
  int b = blockIdx.x;
  sup[t] = 0;
  ord[t] = order[(size_t)b * N + t];
  __syncthreads();
  size_t base = (size_t)b * N * N;
  for (int i = 0; i < N; ++i) {
    int ki = sup[i] ? 0 : 1;                 // final by end of iter i-1
    if (ki && t > i) {
      if (iou[base + (size_t)ord[i] * N + ord[t]] > NMS_IOU) sup[t] = 1;
    }
    __syncthreads();
  }
  keep[(size_t)b * N + ord[t]] = sup[t] ? 0 : 1;
}

// ---------------------------------------------------------------------------
// K5: merge GEMM on WMMA f32 16x16x4.
//   D[j,c] = sum_n w[n,j] * Bm[n,c],  w[n,j] = score_n * (iou[n,j] > 0.7)
//   Bm[n, 0..5] = boxes[n, 0..5];  Bm[n, 6] = 1.0  (=> D[:,6] = w.sum(0))
// One wave per 16-row output tile; K unrolled by 8 (2 WMMAs / iteration).
// All loads are unconditional (clamped anchor column, hoisted score loads);
// data selection is pure v_cndmask so the K-loop has no EXEC manipulation
// (WMMA requires EXEC all-ones). Addresses are maintained as bumped pointers.
// VGPR layouts per ISA 7.12.2:
//   A 16x4 : v0 = {K0 | K2}, v1 = {K1 | K3}, M = lane&15
//   B 4x16 : row-striped across lanes; assumed v0 = {K0 | K2}, v1 = {K1 | K3},
//            N = lane&15 (mirror of A's K split — ISA table omits B 4x16)
//   C/D    : VGPR v -> rows v (lanes 0-15) and v+8 (lanes 16-31), N = lane&15
// ---------------------------------------------------------------------------
__global__ void merge_wmma_kernel(const float* __restrict__ anchors,
                                  const float* __restrict__ scores,
                                  const float* __restrict__ iou,
                                  float* __restrict__ raw, int N)
{
  int lane = threadIdx.x;
  int tilesPerBatch = N / 16;
  int b = blockIdx.x / tilesPerBatch;
  int jbase = (blockIdx.x % tilesPerBatch) * 16;

  int m   = lane & 15;
  bool hi = lane >= 16;
  int koffA = hi ? 2 : 0;      // K offsets for A/B VGPR0 ({K0|K2}) ...
  int koffB = hi ? 3 : 1;      // ... and VGPR1 ({K1|K3})
  int col = jbase + m;
  int cclamp = (m < 6) ? m : 6;            // always-valid anchor column
  bool cIsBox = (m < 6);
  float cPad  = (m == 6) ? 1.0f : 0.0f;    // ones column => column sums

  // bumped pointers (no per-iteration 64-bit address math)
  const float* pIa = iou + (size_t)b * N * N + (size_t)koffA * N + col;
  const float* pIb = iou + (size_t)b * N * N + (size_t)koffB * N + col;
  const float* pSa = scores + (size_t)b * N + koffA;
  const float* pSb = scores + (size_t)b * N + koffB;
  const float* pPa = anchors + ((size_t)b * N + koffA) * 7 + cclamp;
  const float* pPb = anchors + ((size_t)b * N + koffB) * 7 + cclamp;

  const size_t strI4 = (size_t)4 * N;   // 4 rows of iou
  const size_t strI8 = (size_t)8 * N;

  v8f acc = {0.f, 0.f, 0.f, 0.f, 0.f, 0.f, 0.f, 0.f};

  for (int k0 = 0; k0 < N; k0 += 8) {
    // --- issue all 12 loads unconditionally (MLP before the wait) ---
    float i0 = pIa[0];
    float i1 = pIb[0];
    float i2 = pIa[strI4];
    float i3 = pIb[strI4];
    float s0 = pSa[0], s1 = pSb[0], s2 = pSa[4], s3 = pSb[4];
    float p0 = pPa[0], p1 = pPb[0], p2 = pPa[28], p3 = pPb[28];

    // prefetch next chunk's iou rows (uniform guard: scalar branch only)
    if (k0 + 8 < N) {
      __builtin_prefetch(pIa + strI8, 0, 0);
      __builtin_prefetch(pIb + strI8, 0, 0);
      __builtin_prefetch(pIa + strI8 + strI4, 0, 0);
      __builtin_prefetch(pIb + strI8 + strI4, 0, 0);
    }

    // --- pure-select operand construction (no divergent branches) ---
    v2f A0 = { (i0 > MERGE_IOU) ? s0 : 0.0f, (i1 > MERGE_IOU) ? s1 : 0.0f };
    v2f B0 = { cIsBox ? p0 : cPad, cIsBox ? p1 : cPad };
    acc = __builtin_amdgcn_wmma_f32_16x16x4_f32(
        false, A0, false, B0, (short)0, acc, false, false);

    v2f A1 = { (i2 > MERGE_IOU) ? s2 : 0.0f, (i3 > MERGE_IOU) ? s3 : 0.0f };
    v2f B1 = { cIsBox ? p2 : cPad, cIsBox ? p3 : cPad };
    acc = __builtin_amdgcn_wmma_f32_16x16x4_f32(
        false, A1, false, B1, (short)0, acc, false, false);

    pIa += strI8; pIb += strI8;
    pSa += 8;     pSb += 8;
    pPa += 56;    pPb += 56;
  }

#pragma unroll
  for (int v = 0; v < 8; ++v) {
    int j = jbase + v + (hi ? 8 : 0);
    if (m < 7) raw[((size_t)b * N + j) * 8 + m] = acc[v];
  }
}

// ---------------------------------------------------------------------------
// K6: normalize by column sum (raw[:,6]), angle mod 2pi, apply keep mask.
// ---------------------------------------------------------------------------
__global__ void finalize_kernel(const float* __restrict__ anchors,
                                const float* __restrict__ raw,
                                const int* __restrict__ keep,
                                float* __restrict__ out, int B, int N)
{
  int idx = blockIdx.x * blockDim.x + threadIdx.x;
  int total = B * N * 7;
  if (idx >= total) return;
  int c = idx % 7;
  int j = (idx / 7) % N;
  int b = idx / (7 * N);
  float kf = keep[(size_t)b * N + j] ? 1.0f : 0.0f;
  float val;
  if (c < 6) {
    float den = fmaxf(raw[((size_t)b * N + j) * 8 + 6], FEPS);
    val = raw[((size_t)b * N + j) * 8 + c] / den;
  } else {
    float ry = anchors[((size_t)b * N + j) * 7 + 6];
    float r = fmodf(ry, TWO_PI_F);
    if (r < 0.f) r += TWO_PI_F;
    val = r;
  }
  out[idx] = val * kf;
}

// ---------------------------------------------------------------------------
extern "C" void kernel_launch(void* const* d_in, const int* in_sizes, int n_in,
                              void* d_out, int out_size, void* d_ws, size_t ws_size,
                              hipStream_t stream)
{
  const float* anchors = (const float*)d_in[0];   // (B, N, 7)
  const float* cls     = (const float*)d_in[1];   // (B, N)
  float* out = (float*)d_out;                     // (B, N, 7)

  const int N = NBOX;
  const int B = in_sizes[0] / (N * 7);

  // workspace carve-up (all 4-byte types)
  float* iou    = (float*)d_ws;                              // B*N*N
  float* scores = iou + (size_t)B * N * N;                   // B*N
  int*   order  = (int*)(scores + (size_t)B * N);            // B*N
  int*   keep   = order + (size_t)B * N;                     // B*N
  float* raw    = (float*)(keep + (size_t)B * N);            // B*N*8

  {
    int n = B * N;
    score_kernel<<<(n + 255) / 256, 256, 0, stream>>>(cls, scores, n);
  }
  {
    size_t total = (size_t)B * N * N;
    int blocks = (int)((total + 255) / 256);
    iou_kernel<<<blocks, 256, 0, stream>>>(anchors, iou, B, N);
  }
  sort_kernel<<<B, N, 0, stream>>>(scores, order, N);
  nms_kernel<<<B, N, 0, stream>>>(iou, order, keep, N);
  merge_wmma_kernel<<<B * (N / 16), 32, 0, stream>>>(anchors, scores, iou, raw, N);
  {
    int n = B * N * 7;
    finalize_kernel<<<(n + 255) / 256, 256, 0, stream>>>(anchors, raw, keep, out, B, N);
  }
}